// AlignmentContrastiveLoss_55808805044518
// MI455X (gfx1250) — compile-verified
//
#include <hip/hip_runtime.h>
#include <math.h>

typedef float v2f __attribute__((ext_vector_type(2)));
typedef float v8f __attribute__((ext_vector_type(8)));

#define D_DIM 512
#define TILE 16

// ---------------------------------------------------------------- accumulators
// acc[0]=pos_sum acc[1]=pos_cnt acc[2]=neg_sum acc[3]=neg_cnt
__global__ void init_acc_kernel(float* __restrict__ acc) {
  if (threadIdx.x < 4) acc[threadIdx.x] = 0.0f;
}

// ------------------------------------------------------------- L2 normalization
// One 256-thread block per row (512 floats, 2 per thread).
__global__ void normalize_kernel(const float* __restrict__ emb,
                                 float* __restrict__ e) {
  __shared__ float red[8];
  const int row = blockIdx.x;
  const int t = threadIdx.x;
  const size_t base = (size_t)row * D_DIM;
  const float x = emb[base + t];
  const float y = emb[base + t + 256];
  float ss = x * x + y * y;
  for (int off = 16; off > 0; off >>= 1) ss += __shfl_down(ss, off, 32);
  if ((t & 31) == 0) red[t >> 5] = ss;
  __syncthreads();
  const float tot = red[0] + red[1] + red[2] + red[3] +
                    red[4] + red[5] + red[6] + red[7];
  const float scale = 1.0f / fmaxf(sqrtf(tot), 1e-12f);
  e[base + t] = x * scale;
  e[base + t + 256] = y * scale;
}

// -------------------------------------------- fused masked WMMA GEMM-reduction
// One wave per 16x16 tile of sims = e @ e^T (upper-triangular tiles only).
// V_WMMA_F32_16X16X4_F32 operand layout (ISA 7.12.2):
//   A 16x4 : lane -> M = lane&15 ; VGPR{0,1} -> K = 2*(lane>>4) + {0,1}
//   B 4x16 : lane -> N = lane&15 ; VGPR{0,1} -> K = 2*(lane>>4) + {0,1}
//   C 16x16: VGPR r, lane -> (M = r + 8*(lane>>4), N = lane&15)
// => both operands are contiguous float2 loads from the row-major e matrix.
__global__ void pos_tile_kernel(const float* __restrict__ e,
                                const int* __restrict__ labels,
                                const int* __restrict__ graphs,
                                const int* __restrict__ cats,
                                float* __restrict__ acc) {
  const int lane = threadIdx.x & 31;
  const int wave = threadIdx.x >> 5;
  const int tj = blockIdx.x;
  const int ti = blockIdx.y * 8 + wave;
  if (ti > tj) return;  // wave-uniform: strictly-lower tiles contribute nothing

  const int i_base = ti * TILE;
  const int j_base = tj * TILE;
  const int nlo = lane & 15;
  const int hi = lane >> 4;

  // Validity mask for the 8 C elements this lane owns:
  //   element r -> (i = i_base + r + 8*hi, j = j_base + nlo)
  const int j = j_base + nlo;
  const int labj = labels[j];
  const int gj = graphs[j];
  const int cj = cats[j] < 3;
  unsigned m8 = 0;
#pragma unroll
  for (int r = 0; r < 8; ++r) {
    const int i = i_base + r + 8 * hi;
    const int v = (i < j) & (labels[i] == labj) & (graphs[i] != gj) &
                  (cats[i] < 3) & cj;
    m8 |= (unsigned)v << r;
  }
  // ~89% of tiles have no valid pair under uniform labels -> skip the K-loop.
  if (__ballot(m8 != 0) == 0ull) return;  // wave-uniform, EXEC stays all-1s

  const float* ap = e + (size_t)(i_base + nlo) * D_DIM + 2 * hi;
  const float* bp = e + (size_t)(j_base + nlo) * D_DIM + 2 * hi;
  v8f c = {};
#pragma unroll 4
  for (int k = 0; k < D_DIM; k += 4) {
    const v2f a = *(const v2f*)(ap + k);
    const v2f b = *(const v2f*)(bp + k);
    // 8 args: (neg_a, A, neg_b, B, c_mod, C, reuse_a, reuse_b)
    c = __builtin_amdgcn_wmma_f32_16x16x4_f32(false, a, false, b, (short)0, c,
                                              false, false);
  }

  float lsum = 0.0f, lcnt = 0.0f;
#pragma unroll
  for (int r = 0; r < 8; ++r) {
    if ((m8 >> r) & 1u) {
      lsum += 1.0f - c[r];
      lcnt += 1.0f;
    }
  }
  for (int off = 16; off > 0; off >>= 1) {
    lsum += __shfl_down(lsum, off, 32);
    lcnt += __shfl_down(lcnt, off, 32);
  }
  if (lane == 0) {
    atomicAdd(&acc[0], lsum);
    atomicAdd(&acc[1], lcnt);
  }
}

// --------------------------------------------------------- negative-pair loss
// One wave per sampled pair: 512-d dot + mask. 2.5 MFLOP total — trivial.
__global__ void neg_kernel(const float* __restrict__ e,
                           const int* __restrict__ labels,
                           const int* __restrict__ graphs,
                           const int* __restrict__ cats,
                           const int* __restrict__ idx1,
                           const int* __restrict__ idx2,
                           int S, float* __restrict__ acc) {
  const int lane = threadIdx.x & 31;
  const int p = blockIdx.x * 8 + (threadIdx.x >> 5);
  if (p >= S) return;
  const int i1 = idx1[p];
  const int i2 = idx2[p];
  const float* r1 = e + (size_t)i1 * D_DIM;
  const float* r2 = e + (size_t)i2 * D_DIM;
  float dot = 0.0f;
#pragma unroll
  for (int t = 0; t < D_DIM / 32; ++t)
    dot += r1[lane + 32 * t] * r2[lane + 32 * t];
  for (int off = 16; off > 0; off >>= 1) dot += __shfl_down(dot, off, 32);
  if (lane == 0) {
    const bool valid = (graphs[i1] != graphs[i2]) &&
                       (labels[i1] != labels[i2]) &&
                       ((cats[i1] < 3) || (cats[i2] < 3));
    if (valid) {
      atomicAdd(&acc[2], fmaxf(dot, 0.0f));  // MARGIN == 0
      atomicAdd(&acc[3], 1.0f);
    }
  }
}

// ---------------------------------------------------------------------- finish
__global__ void finalize_kernel(const float* __restrict__ acc,
                                float* __restrict__ out) {
  if (threadIdx.x == 0) {
    const float pos = acc[1] > 0.0f ? acc[0] / acc[1] : 0.0f;
    const float neg = acc[3] > 0.0f ? acc[2] / acc[3] : 0.0f;
    out[0] = pos + neg;
  }
}

extern "C" void kernel_launch(void* const* d_in, const int* in_sizes, int n_in,
                              void* d_out, int out_size, void* d_ws,
                              size_t ws_size, hipStream_t stream) {
  const float* emb = (const float*)d_in[0];
  const int* labels = (const int*)d_in[1];
  const int* graphs = (const int*)d_in[2];
  const int* cats = (const int*)d_in[3];
  const int* idx1 = (const int*)d_in[4];
  const int* idx2 = (const int*)d_in[5];
  const int N = in_sizes[1];  // 8192
  const int S = in_sizes[4];  // 5000

  float* e = (float*)d_ws;                 // N*D normalized rows (16 MB)
  float* acc = e + (size_t)N * D_DIM;      // 4 scalar accumulators
  float* out = (float*)d_out;

  init_acc_kernel<<<1, 32, 0, stream>>>(acc);
  normalize_kernel<<<N, 256, 0, stream>>>(emb, e);

  const int T = N / TILE;  // 512 tiles per dim
  pos_tile_kernel<<<dim3(T, T / 8), 256, 0, stream>>>(e, labels, graphs, cats,
                                                      acc);
  neg_kernel<<<(S + 7) / 8, 256, 0, stream>>>(e, labels, graphs, cats, idx1,
                                              idx2, S, acc);
  finalize_kernel<<<1, 32, 0, stream>>>(acc, out);
}